// AdjMP_69329362092561
// MI455X (gfx1250) — compile-verified
//
#include <hip/hip_runtime.h>

// ---------------------------------------------------------------------------
// GCN-style 4-layer SpMM:  x_{k+1} = D^-1/2 (A + I) D^-1/2 x_k
// N nodes, D=128 f32 features, E int32 edges.
// Strategy: build CSR once, then 4 atomic-free gather layers, one wave32 per
// destination node, lane = float4 of the feature row. Adjacency segments are
// double-buffered into LDS with CDNA5 async global->LDS loads (ASYNCcnt).
// ---------------------------------------------------------------------------

#define FEAT4 32            // 128 features / 4 per lane
#define WPB   8             // waves per block (256 threads)

__device__ __forceinline__ void async_ld_b32(unsigned lds_addr, const void* gaddr) {
  // CDNA5: GLOBAL_LOAD_ASYNC_TO_LDS_B32  (vdst = LDS byte address, vaddr = 64b global)
  asm volatile("global_load_async_to_lds_b32 %0, %1, off"
               :: "v"(lds_addr), "v"(gaddr)
               : "memory");
}
__device__ __forceinline__ void wait_async_le2() {
  asm volatile("s_wait_asynccnt 2" ::: "memory");
}
__device__ __forceinline__ void wait_async_0() {
  asm volatile("s_wait_asynccnt 0" ::: "memory");
}

// ---------------- degree / CSR construction ----------------

__global__ void adjmp_zero_deg(int* __restrict__ degi, int N) {
  int i = blockIdx.x * blockDim.x + threadIdx.x;
  if (i < N) degi[i] = 0;
}

__global__ void adjmp_count(const int* __restrict__ rows, int* __restrict__ degi, int E) {
  int e = blockIdx.x * blockDim.x + threadIdx.x;
  if (e < E) atomicAdd(&degi[rows[e]], 1);
}

// single-block scan: off[i] = exclusive prefix (off[0]=0, off[N]=E), cursor = copy
__global__ void adjmp_scan(const int* __restrict__ degi, int* __restrict__ off,
                           int* __restrict__ cursor, int N) {
  __shared__ int sh[1024];
  __shared__ int carry_s;
  if (threadIdx.x == 0) carry_s = 0;
  __syncthreads();
  for (int base = 0; base < N; base += 1024) {
    int i = base + (int)threadIdx.x;
    int v = (i < N) ? degi[i] : 0;
    sh[threadIdx.x] = v;
    __syncthreads();
    for (int d = 1; d < 1024; d <<= 1) {
      int t = (threadIdx.x >= (unsigned)d) ? sh[threadIdx.x - d] : 0;
      __syncthreads();
      sh[threadIdx.x] += t;
      __syncthreads();
    }
    int incl = sh[threadIdx.x];
    int c = carry_s;
    if (i < N) {
      off[i + 1]  = c + incl;       // inclusive prefix -> off[i+1]
      cursor[i]   = c + incl - v;   // exclusive prefix = row start
    }
    __syncthreads();
    if (threadIdx.x == 1023) carry_s = c + sh[1023];
    __syncthreads();
  }
  if (threadIdx.x == 0) off[0] = 0;
}

__global__ void adjmp_dinv(const int* __restrict__ degi, float* __restrict__ dinv,
                           float* __restrict__ wi, int N) {
  int i = blockIdx.x * blockDim.x + threadIdx.x;
  if (i < N) {
    float d = (float)(degi[i] + 1);          // +1 self loop; deg counted over rows
    float r = 1.0f / sqrtf(d);
    dinv[i] = r;
    wi[i]   = r * r;                         // self-loop weight dinv[i]*dinv[i]
  }
}

__global__ void adjmp_scatter(const int* __restrict__ rows, const int* __restrict__ cols,
                              const float* __restrict__ dinv, int* __restrict__ cursor,
                              int* __restrict__ csr_col, float* __restrict__ csr_w, int E) {
  int e = blockIdx.x * blockDim.x + threadIdx.x;
  if (e < E) {
    int r = rows[e];
    int c = cols[e];
    int pos = atomicAdd(&cursor[r], 1);
    csr_col[pos] = c;
    csr_w[pos]   = dinv[r] * dinv[c];
  }
}

// ---------------- SpMM layer: one wave per destination node ----------------
// LDS layout per (wave, buffer): 32 ints col | 32 floats w  (64 dwords)

__global__ void __launch_bounds__(256)
adjmp_spmm_csr(const int* __restrict__ off, const int* __restrict__ csr_col,
               const float* __restrict__ csr_w, const float* __restrict__ wi,
               const float* __restrict__ h, float* __restrict__ out, int N) {
  __shared__ int smem[WPB * 2 * 64];

  const int lane = threadIdx.x & 31;
  const int wave = threadIdx.x >> 5;
  const int stride = gridDim.x * WPB;
  const float4* __restrict__ h4 = (const float4*)h;
  float4* __restrict__ o4 = (float4*)out;

  // stage node i's first <=32 edges into buffer b (always exactly 2 async ops;
  // csr arrays are padded by 32 entries so unclamped lane reads stay in bounds)
  auto stage = [&](int b, int i) {
    int e0 = off[i];
    unsigned lb = (unsigned)(size_t)&smem[(wave * 2 + b) * 64];
    async_ld_b32(lb + (unsigned)(lane * 4),       (const void*)(csr_col + e0 + lane));
    async_ld_b32(lb + 128u + (unsigned)(lane * 4), (const void*)(csr_w  + e0 + lane));
  };

  int i = blockIdx.x * WPB + wave;
  int b = 0;
  if (i < N) stage(0, i);

  while (i < N) {
    int inext = i + stride;
    if (inext < N) {
      stage(b ^ 1, inext);   // 4 outstanding: wait until only next-node's 2 remain
      wait_async_le2();
    } else {
      wait_async_0();
    }

    int e0 = off[i];
    int e1 = off[i + 1];
    int len = e1 - e0;
    int lcap = len < 32 ? len : 32;

    const int*   lcol = &smem[(wave * 2 + b) * 64];
    const float* lw   = (const float*)(lcol + 32);

    // self-loop term initializes the accumulator
    float s = wi[i];
    float4 v0 = h4[i * FEAT4 + lane];
    float4 acc = make_float4(s * v0.x, s * v0.y, s * v0.z, s * v0.w);

    for (int j = 0; j < lcap; ++j) {
      int   c  = lcol[j];                       // wave-uniform LDS broadcast
      float wt = lw[j];
      if (j + 1 < lcap)
        __builtin_prefetch(&h4[lcol[j + 1] * FEAT4 + lane], 0, 1);
      float4 v = h4[c * FEAT4 + lane];
      acc.x += wt * v.x; acc.y += wt * v.y;
      acc.z += wt * v.z; acc.w += wt * v.w;
    }
    // rare overflow (degree > 32): direct uniform loads
    for (int e = e0 + 32; e < e1; ++e) {
      int   c  = csr_col[e];
      float wt = csr_w[e];
      float4 v = h4[c * FEAT4 + lane];
      acc.x += wt * v.x; acc.y += wt * v.y;
      acc.z += wt * v.z; acc.w += wt * v.w;
    }

    o4[i * FEAT4 + lane] = acc;
    b ^= 1;
    i = inext;
  }
}

// ---------------------------------------------------------------------------

extern "C" void kernel_launch(void* const* d_in, const int* in_sizes, int n_in,
                              void* d_out, int out_size, void* d_ws, size_t ws_size,
                              hipStream_t stream) {
  (void)n_in; (void)out_size; (void)ws_size;
  const float* x  = (const float*)d_in[0];
  const int*   ei = (const int*)d_in[1];           // int32 (JAX x64 disabled)
  const int N = in_sizes[0] / 128;
  const int E = in_sizes[1] / 2;
  const int* rows = ei;
  const int* cols = ei + E;

  // workspace carve-up (4-byte units, each region 16B aligned)
  auto a4 = [](size_t n) { return (n + 3) & ~(size_t)3; };
  float* ws = (float*)d_ws;
  size_t o = 0;
  int*   degi    = (int*)(ws + o);   o += a4((size_t)N);
  int*   off     = (int*)(ws + o);   o += a4((size_t)N + 1);
  int*   cursor  = (int*)(ws + o);   o += a4((size_t)N);
  float* dinv    = ws + o;           o += a4((size_t)N);
  float* wi      = ws + o;           o += a4((size_t)N);
  int*   csr_col = (int*)(ws + o);   o += a4((size_t)E + 32);   // +32 pad for async tiles
  float* csr_w   = ws + o;           o += a4((size_t)E + 32);
  float* bufA    = ws + o;           // N*128 floats
  float* out     = (float*)d_out;

  dim3 blk(256);
  int gN = (N + 255) / 256;
  int gE = (E + 255) / 256;

  // one-time CSR build
  adjmp_zero_deg<<<gN, blk, 0, stream>>>(degi, N);
  adjmp_count  <<<gE, blk, 0, stream>>>(rows, degi, E);
  adjmp_scan   <<<1, 1024, 0, stream>>>(degi, off, cursor, N);
  adjmp_dinv   <<<gN, blk, 0, stream>>>(degi, dinv, wi, N);
  adjmp_scatter<<<gE, blk, 0, stream>>>(rows, cols, dinv, cursor, csr_col, csr_w, E);

  // 4 layers, ping-pong x -> bufA -> out -> bufA -> out
  int waves_needed = N;
  int blocks = (waves_needed + WPB - 1) / WPB;
  if (blocks > 2048) blocks = 2048;              // ~3 nodes/wave -> async pipelining
  const float* hsrc = x;
  for (int l = 0; l < 4; ++l) {
    float* dst = (l & 1) ? out : bufA;
    adjmp_spmm_csr<<<blocks, blk, 0, stream>>>(off, csr_col, csr_w, wi, hsrc, dst, N);
    hsrc = dst;
  }
}